// HomeostaticField_25615184953595
// MI455X (gfx1250) — compile-verified
//
#include <hip/hip_runtime.h>

typedef float v2f __attribute__((ext_vector_type(2)));
typedef float v8f __attribute__((ext_vector_type(8)));

#define SDIM  512
#define DPAD  513            // SDIM + 1
#define NA    16
#define ALPHA 0.1f
#define HEPS  1e-7f
#define WAVES 8
#define ROWS_PER_BLOCK (WAVES * 16)

__global__ __launch_bounds__(256)
void HomeostaticField_kernel(const float* __restrict__ x,
                             const float* __restrict__ anchors,
                             float* __restrict__ out)
{
    // LDS: ~77 KB total (gfx1250 WGP has 320 KB)
    __shared__ float afT[516 * 16];        // B-matrix layout: af'[k][a], time +, spatial -, k=513..515 zero-pad
    __shared__ float af2[NA * DPAD];       // af row-major (true values) for epilogue, conflict-free reads
    __shared__ float thetaS[WAVES * 16 * 17];
    __shared__ float red[16 * 17];
    __shared__ float tvals[NA];
    __shared__ float rowP[WAVES * 16];
    __shared__ float rowQ[WAVES * 16];
    __shared__ int   rowI[WAVES * 16];

    const int tid  = threadIdx.x;
    const int wave = tid >> 5;
    const int lane = tid & 31;
    const int m    = lane & 15;     // row (A) / col (B) index within tile
    const int hi   = lane >> 4;     // half-wave
    const int koff = hi << 1;       // K sub-offset: 0 or 2

    // ---------- stage anchors: recompute time component (_proj), build af' and af ----------
    {
        int a = tid >> 4;
        int j = tid & 15;
        float acc = 0.f;
        for (int d = 1 + j; d < DPAD; d += 16) {
            float v = anchors[a * DPAD + d];
            acc = fmaf(v, v, acc);
        }
        red[a * 17 + j] = acc;
    }
    __syncthreads();
    if (tid < NA) {
        float s = 0.f;
        for (int j = 0; j < 16; ++j) s += red[tid * 17 + j];
        tvals[tid] = sqrtf(1.0f + s);
    }
    __syncthreads();
    for (int a = 0; a < NA; ++a) {
        float t = tvals[a];
        for (int d = tid; d < DPAD; d += 256) {
            float v  = anchors[a * DPAD + d];
            af2[a * DPAD + d] = (d == 0) ? t : v;
            afT[d * 16 + a]   = (d == 0) ? t : -v;   // fold Minkowski metric into B
        }
    }
    if (tid < 48) afT[513 * 16 + tid] = 0.f;         // zero-pad K=513..515
    __syncthreads();

    // ---------- WMMA: theta tile = X(16xK) * af'(Kx16), full fp32 ----------
    const long  rowBase = (long)blockIdx.x * ROWS_PER_BLOCK + wave * 16;
    const float* xr = x + (size_t)(rowBase + m) * DPAD + koff;   // A-frag per-lane base

    v8f acc0 = {}; v8f acc1 = {}; v8f acc2 = {}; v8f acc3 = {};
    for (int k0 = 0; k0 < SDIM; k0 += 16) {
        // 4 independent accumulators break the WMMA RAW dependency chain
        {
            v2f av; av.x = xr[k0 + 0]; av.y = xr[k0 + 1];
            int kb = k0 + koff;
            v2f bv; bv.x = afT[kb * 16 + m]; bv.y = afT[(kb + 1) * 16 + m];
            acc0 = __builtin_amdgcn_wmma_f32_16x16x4_f32(false, av, false, bv, (short)0, acc0, false, false);
        }
        {
            v2f av; av.x = xr[k0 + 4]; av.y = xr[k0 + 5];
            int kb = k0 + 4 + koff;
            v2f bv; bv.x = afT[kb * 16 + m]; bv.y = afT[(kb + 1) * 16 + m];
            acc1 = __builtin_amdgcn_wmma_f32_16x16x4_f32(false, av, false, bv, (short)0, acc1, false, false);
        }
        {
            v2f av; av.x = xr[k0 + 8]; av.y = xr[k0 + 9];
            int kb = k0 + 8 + koff;
            v2f bv; bv.x = afT[kb * 16 + m]; bv.y = afT[(kb + 1) * 16 + m];
            acc2 = __builtin_amdgcn_wmma_f32_16x16x4_f32(false, av, false, bv, (short)0, acc2, false, false);
        }
        {
            v2f av; av.x = xr[k0 + 12]; av.y = xr[k0 + 13];
            int kb = k0 + 12 + koff;
            v2f bv; bv.x = afT[kb * 16 + m]; bv.y = afT[(kb + 1) * 16 + m];
            acc3 = __builtin_amdgcn_wmma_f32_16x16x4_f32(false, av, false, bv, (short)0, acc3, false, false);
        }
    }
    v8f acc = (acc0 + acc1) + (acc2 + acc3);

    // dump theta tile: VGPR r -> tile row (r + 8*hi), col m
    #pragma unroll
    for (int r = 0; r < 8; ++r)
        thetaS[wave * 272 + (r + 8 * hi) * 17 + m] = acc[r];
    __syncthreads();

    // ---------- per-row scalars: K=512 tail, clip, first-occurrence argmin, transcendentals ----------
    if (lane < 16) {
        int rr = lane;
        float xt = x[(size_t)(rowBase + rr) * DPAD + SDIM];   // x[row][512]
        float bestC = 3.4e38f, bestRaw = 0.f;
        int bestI = 0;
        for (int a = 0; a < NA; ++a) {
            float th = fmaf(xt, afT[SDIM * 16 + a], thetaS[wave * 272 + rr * 17 + a]);
            float cl = fmaxf(th, 1.0f + HEPS);
            if (cl < bestC) { bestC = cl; bestRaw = th; bestI = a; }
        }
        float th  = bestC;                       // clipped theta at argmin
        float sq  = sqrtf(th * th - 1.0f);
        float arc = logf(th + sq);               // arccosh(th)
        float c   = arc / sq;
        float s   = ALPHA * c;
        // linner(w,w) = 2*th*th_raw - th^2 - 1  (w = y - th*x, hyperboloid identity)
        float q   = 2.0f * th * bestRaw - th * th - 1.0f;
        float vv  = fmaxf(s * s * q, HEPS);
        float vn  = sqrtf(vv);
        float ch  = coshf(vn);
        float sr  = sinhf(vn) / vn;
        rowP[wave * 16 + rr] = ch - sr * s * th; // coefficient of x
        rowQ[wave * 16 + rr] = sr * s;           // coefficient of y
        rowI[wave * 16 + rr] = bestI;
    }
    __syncthreads();

    // ---------- epilogue: out = P*x + Q*y  (x re-read hits L2; stores coalesced) ----------
    for (int rr = 0; rr < 16; ++rr) {
        float P = rowP[wave * 16 + rr];
        float Q = rowQ[wave * 16 + rr];
        int idx = rowI[wave * 16 + rr];
        const float* xrow = x   + (size_t)(rowBase + rr) * DPAD;
        float*       orow = out + (size_t)(rowBase + rr) * DPAD;
        const float* yrow = af2 + idx * DPAD;
        for (int d = lane; d < DPAD; d += 32) {
            orow[d] = fmaf(P, xrow[d], Q * yrow[d]);
        }
    }
}

extern "C" void kernel_launch(void* const* d_in, const int* in_sizes, int n_in,
                              void* d_out, int out_size, void* d_ws, size_t ws_size,
                              hipStream_t stream) {
    const float* x       = (const float*)d_in[0];
    const float* anchors = (const float*)d_in[1];
    float*       out     = (float*)d_out;
    int rows   = in_sizes[0] / DPAD;             // 131072
    int blocks = rows / ROWS_PER_BLOCK;          // 1024 (exact for B=131072)
    hipLaunchKernelGGL(HomeostaticField_kernel, dim3(blocks), dim3(256), 0, stream,
                       x, anchors, out);
}